// EltwiseSubEmbed_61735859913133
// MI455X (gfx1250) — compile-verified
//
#include <hip/hip_runtime.h>

#define NP 128
#define NG 256
#define NC 2048
#define HW 64   // 8*8 spatial

typedef __attribute__((ext_vector_type(2))) float v2f;
typedef __attribute__((ext_vector_type(4))) float v4f;
typedef __attribute__((ext_vector_type(8))) float v8f;

// ---------------------------------------------------------------------------
// Kernel 1: global average pool. Treat probe+gallery as (NP+NG)*NC rows of 64
// contiguous floats. One wave32 = 2 rows: lanes 0-15 row A, lanes 16-31 row B,
// each lane loads one float4 -> one contiguous, fully coalesced 512B burst per
// wave. 16-lane xor-shuffle reduce, lane sub==0 writes the mean.
// ---------------------------------------------------------------------------
__global__ void pool_kernel(const float* __restrict__ probe,
                            const float* __restrict__ gallery,
                            float* __restrict__ pmean,
                            float* __restrict__ gmean) {
  const int lane = threadIdx.x & 31;
  const int wave = threadIdx.x >> 5;
  const int sub  = lane & 15;   // position within the row (float4 granularity)
  const int half = lane >> 4;   // which of the wave's two rows

  long long row = (long long)blockIdx.x * 16 + (long long)wave * 2 + half;
  const long long n_probe_rows = (long long)NP * NC;

  const float* src;
  float* dst;
  if (row < n_probe_rows) {
    src = probe + row * HW;
    dst = pmean + row;
  } else {
    long long r2 = row - n_probe_rows;
    src = gallery + r2 * HW;
    dst = gmean + r2;
  }

  v4f v = *(const v4f*)(src + sub * 4);
  float s = v.x + v.y + v.z + v.w;
  // reduce across the 16 lanes of this half (xor of bits 0..3 stays in-half)
  s += __shfl_xor(s, 1, 32);
  s += __shfl_xor(s, 2, 32);
  s += __shfl_xor(s, 4, 32);
  s += __shfl_xor(s, 8, 32);
  if (sub == 0) *dst = s * (1.0f / 64.0f);
}

// ---------------------------------------------------------------------------
// Kernel 2: squared L2 norms of the pooled rows. One wave per row, coalesced
// stride-32 loads, full 32-lane reduce.
// ---------------------------------------------------------------------------
__global__ void norm_kernel(const float* __restrict__ pmean,
                            const float* __restrict__ gmean,
                            float* __restrict__ pn,
                            float* __restrict__ gn) {
  const int lane = threadIdx.x & 31;
  const int wave = threadIdx.x >> 5;
  const int row  = blockIdx.x * 8 + wave;   // 8 waves / 256-thread block

  const float* src;
  float* dst;
  if (row < NP) {
    src = pmean + (long long)row * NC;
    dst = pn + row;
  } else {
    src = gmean + (long long)(row - NP) * NC;
    dst = gn + (row - NP);
  }

  float s = 0.0f;
  #pragma unroll 4
  for (int i = lane; i < NC; i += 32) {
    float x = src[i];
    s += x * x;
  }
  s += __shfl_xor(s, 1, 32);
  s += __shfl_xor(s, 2, 32);
  s += __shfl_xor(s, 4, 32);
  s += __shfl_xor(s, 8, 32);
  s += __shfl_xor(s, 16, 32);
  if (lane == 0) *dst = s;
}

// ---------------------------------------------------------------------------
// Kernel 3: D[i,j] = pn[i] + gn[j] - 2 * (P @ G^T)[i,j] via fp32 WMMA.
// One wave per 16x16 tile; blockDim = 32 so EXEC is all-ones (WMMA req).
//
// A (16x4 f32) VGPR layout: lanes 0-15 hold M=lane, K={k0,k0+1};
// lanes 16-31 hold M=lane-16, K={k0+2,k0+3}  -> per-lane float2 load at
// row m, offset k0 + 2*(lane>=16). B = G^T mirrors this with gallery rows.
// C/D layout: VGPR r holds M = r + 8*(lane>=16), N = lane%16.
// ---------------------------------------------------------------------------
__global__ void dist_wmma_kernel(const float* __restrict__ pmean,
                                 const float* __restrict__ gmean,
                                 const float* __restrict__ pn,
                                 const float* __restrict__ gn,
                                 float* __restrict__ out) {
  const int lane = threadIdx.x;     // blockDim.x == 32
  const int n16  = lane & 15;
  const int hi   = lane >> 4;       // 0 -> K pair {k0,k0+1}, 1 -> {k0+2,k0+3}
  const int ti   = blockIdx.y;      // probe tile   (0..7)
  const int tj   = blockIdx.x;      // gallery tile (0..15)

  const float* arow = pmean + (long long)(ti * 16 + n16) * NC + 2 * hi;
  const float* brow = gmean + (long long)(tj * 16 + n16) * NC + 2 * hi;

  v8f acc = {};
  #pragma unroll 8
  for (int k = 0; k < NC; k += 4) {
    v2f a = *(const v2f*)(arow + k);
    v2f b = *(const v2f*)(brow + k);
    // 8 args: (neg_a, A, neg_b, B, c_mod, C, reuse_a, reuse_b)
    acc = __builtin_amdgcn_wmma_f32_16x16x4_f32(
        /*neg_a=*/false, a, /*neg_b=*/false, b,
        /*c_mod=*/(short)0, acc, /*reuse_a=*/false, /*reuse_b=*/false);
  }

  const int n = tj * 16 + n16;
  const float gnv = gn[n];
  #pragma unroll
  for (int r = 0; r < 8; ++r) {
    const int m = ti * 16 + hi * 8 + r;
    out[m * NG + n] = pn[m] + gnv - 2.0f * acc[r];
  }
}

extern "C" void kernel_launch(void* const* d_in, const int* in_sizes, int n_in,
                              void* d_out, int out_size, void* d_ws, size_t ws_size,
                              hipStream_t stream) {
  const float* probe   = (const float*)d_in[0];  // [128, 2048, 8, 8] f32
  const float* gallery = (const float*)d_in[1];  // [256, 2048, 8, 8] f32
  float* out = (float*)d_out;                    // [128*256] f32

  // Workspace layout (floats): pmean | gmean | pn | gn  (~3.004 MB)
  float* ws    = (float*)d_ws;
  float* pmean = ws;                       // NP*NC
  float* gmean = pmean + (size_t)NP * NC;  // NG*NC
  float* pn    = gmean + (size_t)NG * NC;  // NP
  float* gn    = pn + NP;                  // NG

  // Pool: (NP+NG)*NC rows, 16 rows per 256-thread block.
  const int total_rows = (NP + NG) * NC;   // 786432
  pool_kernel<<<total_rows / 16, 256, 0, stream>>>(probe, gallery, pmean, gmean);

  // Norms: 384 rows, 8 waves per block.
  norm_kernel<<<(NP + NG) / 8, 256, 0, stream>>>(pmean, gmean, pn, gn);

  // Distance: 8x16 tiles of 16x16, one wave each.
  dist_wmma_kernel<<<dim3(NG / 16, NP / 16), 32, 0, stream>>>(pmean, gmean, pn, gn, out);
}